// EPPCore_3745211482315
// MI455X (gfx1250) — compile-verified
//
#include <hip/hip_runtime.h>
#include <stdint.h>
#include <stddef.h>

// Problem constants (match reference)
#define BATCH    8
#define HGT      320
#define WID      1024
#define HW       (HGT * WID)          // 327680 pixels per batch
#define MAXINS   200
#define CELLS    9                    // 3x3 matrix per pixel
#define COMP_F   (MAXINS * CELLS)     // 1800 floats per batch comp table

#define THREADS     256               // 8 wave32s
#define TILE_PX     1024              // pixels per tile
#define TILE_F      (TILE_PX * CELLS) // 9216 dwords = 36 KB per tile
#define TILE_V4     (TILE_F / 4)      // 2304 float4
#define V4_PER_THR  (TILE_V4 / THREADS)   // 9 b128 issues per thread per tile
#define PIX_PER_THR (TILE_PX / THREADS)   // 4 pixels per thread per tile
#define BLOCKS_PB   80                // blocks per batch
#define TILES_PB    (HW / (BLOCKS_PB * TILE_PX)) // 4 tiles per block

// ---------- CDNA5 async LDS<->global helpers (gfx1250 ISA, ASYNCcnt) ----------

// Low 32 bits of a generic pointer to LDS == LDS byte offset (AMDGPU addrspacecast layout)
__device__ __forceinline__ uint32_t lds_off(const void* p) {
    return (uint32_t)(uintptr_t)p;
}

// Copies 16 bytes from global address g to LDS offset ldsoff per active lane, tracked by ASYNCcnt
__device__ __forceinline__ void async_load_b128(uint32_t ldsoff, const float* g) {
    asm volatile("global_load_async_to_lds_b128 %0, %1, off"
                 :: "v"(ldsoff), "v"(g) : "memory");
}

// Copies 16 bytes from LDS offset ldsoff to global address g per active lane, tracked by ASYNCcnt
__device__ __forceinline__ void async_store_b128(float* g, uint32_t ldsoff) {
    asm volatile("global_store_async_from_lds_b128 %0, %1, off"
                 :: "v"(g), "v"(ldsoff) : "memory");
}

// 4-byte variant for the odd-sized one-time comp-table load
__device__ __forceinline__ void async_load_b32(uint32_t ldsoff, const float* g) {
    asm volatile("global_load_async_to_lds_b32 %0, %1, off"
                 :: "v"(ldsoff), "v"(g) : "memory");
}

#define WAIT_ASYNC(n) asm volatile("s_wait_asynccnt " #n ::: "memory")

// Hardware f32 atomic add (no return), avoids CAS-loop fallback
__device__ __forceinline__ void global_atomic_add_f32(float* g, float v) {
    asm volatile("global_atomic_add_f32 %0, %1, off"
                 :: "v"(g), "v"(v) : "memory");
}

// ---------------------------- kernels ----------------------------

__global__ __launch_bounds__(THREADS) void epp_zero(float* __restrict__ p, int n) {
    int i = blockIdx.x * blockDim.x + threadIdx.x;
    if (i < n) p[i] = 0.0f;
}

// Phase 1: comp[b, ins, :, :] += 1.1 * compsrc over pixels with instance == ins
__global__ __launch_bounds__(THREADS) void epp_compress(
    const int*   __restrict__ inst,   // [B, 1, H, W]
    const float* __restrict__ src,    // [B, H, W, 3, 3]
    float*       __restrict__ comp)   // [B, MAXINS, 3, 3] (workspace)
{
    __shared__ float lcomp[COMP_F];           // per-block private comp table (7.2 KB)
    __shared__ float stage[2][TILE_F];        // double-buffered input tiles (2 x 36 KB)

    const int t  = threadIdx.x;
    const int b  = blockIdx.y;
    const int p0 = blockIdx.x * (TILES_PB * TILE_PX);

    for (int j = t; j < COMP_F; j += THREADS) lcomp[j] = 0.0f;

    const float* gsrc = src + (size_t)(b * HW + p0) * CELLS;
    const int*   ginst = inst + (size_t)b * HW + p0;

    // Prefetch tile 0: lane t moves float4s t, t+256, ..., t+2048 (coalesced b128)
    {
        const uint32_t s0 = lds_off(&stage[0][0]);
        #pragma unroll
        for (int k = 0; k < V4_PER_THR; ++k)
            async_load_b128(s0 + (uint32_t)(k * THREADS + t) * 16u,
                            gsrc + (k * THREADS + t) * 4);
    }
    __syncthreads();   // lcomp zero-init visible

    for (int T = 0; T < TILES_PB; ++T) {
        const int cur = T & 1;
        if (T + 1 < TILES_PB) {
            // Prefetch next tile into the other buffer, then wait for current tile
            const uint32_t sN = lds_off(&stage[cur ^ 1][0]);
            const float*   gN = gsrc + (size_t)(T + 1) * TILE_F;
            #pragma unroll
            for (int k = 0; k < V4_PER_THR; ++k)
                async_load_b128(sN + (uint32_t)(k * THREADS + t) * 16u,
                                gN + (k * THREADS + t) * 4);
            WAIT_ASYNC(9);   // <=9 outstanding => tile T's 9 loads landed (in-order)
        } else {
            WAIT_ASYNC(0);
        }
        __syncthreads();     // all waves' tile-T data in LDS

        #pragma unroll
        for (int j = 0; j < PIX_PER_THR; ++j) {
            const int q  = j * THREADS + t;                  // pixel within tile
            const int id = ginst[T * TILE_PX + q];
            float v[CELLS];
            #pragma unroll
            for (int k = 0; k < CELLS; ++k)                  // stride-9 reads: conflict-free
                v[k] = stage[cur][q * CELLS + k] * 1.1f;
            #pragma unroll
            for (int k = 0; k < CELLS; ++k)                  // ds_add_f32 LDS atomics
                atomicAdd(&lcomp[id * CELLS + k], v[k]);
        }

        __syncthreads();     // all consumers done before buffer is re-prefetched
    }

    // Flush private table to global comp with hardware f32 atomics
    float* gc = comp + (size_t)b * COMP_F;
    for (int j = t; j < COMP_F; j += THREADS)
        global_atomic_add_f32(gc + j, lcomp[j]);
}

// Phase 2: out[b,h,w,:,:] = comp[b, instance[b,0,h,w], :, :]
__global__ __launch_bounds__(THREADS) void epp_inflate(
    const int*   __restrict__ inst,   // [B, 1, H, W]
    const float* __restrict__ comp,   // [B, MAXINS, 3, 3]
    float*       __restrict__ out)    // [B, H, W, 3, 3]
{
    __shared__ float lcomp[COMP_F];
    __shared__ float ostage[2][TILE_F];       // double-buffered output staging (2 x 36 KB)

    const int t  = threadIdx.x;
    const int b  = blockIdx.y;
    const int p0 = blockIdx.x * (TILES_PB * TILE_PX);

    // Async-load this batch's comp table into LDS (1800 dwords, one-time)
    {
        const float*   gc = comp + (size_t)b * COMP_F;
        const uint32_t c0 = lds_off(&lcomp[0]);
        #pragma unroll
        for (int r = 0; r < (COMP_F + THREADS - 1) / THREADS; ++r) {
            const int j = r * THREADS + t;
            if (j < COMP_F)
                async_load_b32(c0 + (uint32_t)j * 4u, gc + j);
        }
        WAIT_ASYNC(0);
    }
    __syncthreads();

    float*     gout  = out + (size_t)(b * HW + p0) * CELLS;
    const int* ginst = inst + (size_t)b * HW + p0;

    for (int T = 0; T < TILES_PB; ++T) {
        const int cur = T & 1;
        if (T >= 2) {
            WAIT_ASYNC(9);       // stores from tile T-2 (same buffer) drained
            __syncthreads();     // ... for every wave
        }

        #pragma unroll
        for (int j = 0; j < PIX_PER_THR; ++j) {
            const int q  = j * THREADS + t;
            const int id = ginst[T * TILE_PX + q];
            #pragma unroll
            for (int k = 0; k < CELLS; ++k)                  // stride-9 writes: conflict-free
                ostage[cur][q * CELLS + k] = lcomp[id * CELLS + k];
        }

        __syncthreads();         // all ds writes complete before async engine reads LDS

        float*         gT = gout + (size_t)T * TILE_F;
        const uint32_t so = lds_off(&ostage[cur][0]);
        #pragma unroll
        for (int k = 0; k < V4_PER_THR; ++k)                 // coalesced async b128 stores
            async_store_b128(gT + (k * THREADS + t) * 4,
                             so + (uint32_t)(k * THREADS + t) * 16u);
    }
    // s_endpgm performs an implicit wait-idle: remaining async stores drain.
}

// ---------------------------- launch ----------------------------

extern "C" void kernel_launch(void* const* d_in, const int* in_sizes, int n_in,
                              void* d_out, int out_size, void* d_ws, size_t ws_size,
                              hipStream_t stream) {
    (void)in_sizes; (void)n_in; (void)out_size; (void)ws_size;

    const int*   inst = (const int*)d_in[0];     // instance [B,1,H,W] int32
    const float* src  = (const float*)d_in[1];   // compsrc  [B,H,W,3,3] f32
    float*       comp = (float*)d_ws;            // [B,MAXINS,3,3] = 57.6 KB scratch
    float*       out  = (float*)d_out;           // [B,H,W,3,3] f32

    const int compN = BATCH * COMP_F;
    epp_zero<<<dim3((compN + THREADS - 1) / THREADS), dim3(THREADS), 0, stream>>>(comp, compN);

    dim3 grid(BLOCKS_PB, BATCH);
    epp_compress<<<grid, dim3(THREADS), 0, stream>>>(inst, src, comp);
    epp_inflate <<<grid, dim3(THREADS), 0, stream>>>(inst, comp, out);
}